// EvolveGCN_76347338654221
// MI455X (gfx1250) — compile-verified
//
#include <hip/hip_runtime.h>

// ---------------------------------------------------------------------------
// EvolveGCN forward for MI455X (gfx1250, wave32, WMMA).
//  N=100000 nodes, E=1.6M directed edges (3.2M undirected), F_IN=128, H=64.
//  GEMMs use V_WMMA_F32_16X16X4_F32 (fp32 matches reference precision).
//  Propagate = per-edge float4 gather + global_atomic_add_f32 scatter; the
//  whole h + edge_index working set (~51 MB) fits in the 192 MB L2.
// ---------------------------------------------------------------------------

typedef float v2f __attribute__((ext_vector_type(2)));
typedef float v8f __attribute__((ext_vector_type(8)));

#define HID 64

// ---- degree histogram over row = concat(e0,e1) -----------------------------
__global__ __launch_bounds__(256)
void degree_kernel(const long long* __restrict__ e0,
                   const long long* __restrict__ e1,
                   float* __restrict__ deg, long long E) {
  long long i = (long long)blockIdx.x * 256 + threadIdx.x;
  if (i < E) {
    unsafeAtomicAdd(&deg[e0[i]], 1.0f);
    unsafeAtomicAdd(&deg[e1[i]], 1.0f);
  }
}

// ---- deg -> deg^-0.5 in place ---------------------------------------------
__global__ __launch_bounds__(256)
void dinv_kernel(float* __restrict__ deg, int N) {
  int i = blockIdx.x * 256 + threadIdx.x;
  if (i < N) {
    float d = deg[i];
    deg[i] = (d > 0.0f) ? rsqrtf(d) : 0.0f;
  }
}

// ---- C[M,64] = act(A[M,K] @ W[K,64] + bias) via V_WMMA_F32_16X16X4_F32 -----
// Block: 256 threads = 8 waves; wave = (mTile 0..1, nTile 0..3) -> 32x64 tile.
// Requires M % 32 == 0 (true for N=100000) so EXEC is all-ones for WMMA.
template<int K, bool RELU>
__global__ __launch_bounds__(256)
void gemm_bias_kernel(const float* __restrict__ A,
                      const float* __restrict__ W,
                      const float* __restrict__ bias,
                      float* __restrict__ out) {
  // W staged in LDS, pair-interleaved so a B fragment is one ds_load_b64:
  //   lds[((k>>1)*64 + col)*2 + (k&1)] = W[k*64 + col]
  __shared__ float lw[K * HID];
  const int tid = threadIdx.x;
  for (int i = tid; i < K * HID; i += 256) {
    int k = i >> 6, col = i & 63;
    lw[((k >> 1) << 7) + (col << 1) + (k & 1)] = W[i];
  }
  __syncthreads();

  const int lane = tid & 31;
  const int wave = tid >> 5;               // 0..7
  const int m0   = blockIdx.x * 32 + (wave >> 2) * 16;
  const int n0   = (wave & 3) * 16;
  const int half = lane >> 4;              // 0: lanes 0-15, 1: lanes 16-31
  const int l15  = lane & 15;
  const int koff = half * 2;               // ISA A/B layout: K split 0,1 | 2,3
  const float* __restrict__ Ar = A + (size_t)(m0 + l15) * K;

  v8f c = {};
#pragma unroll
  for (int k0 = 0; k0 < K; k0 += 4) {
    const int kk = k0 + koff;                                 // even
    v2f a = *(const v2f*)(Ar + kk);                           // A[row][kk,kk+1]
    v2f b = *(const v2f*)(&lw[((kk >> 1) << 7) + ((n0 + l15) << 1)]); // W[kk..kk+1][col]
    c = __builtin_amdgcn_wmma_f32_16x16x4_f32(
        /*neg_a=*/false, a, /*neg_b=*/false, b,
        /*c_mod=*/(short)0, c, /*reuse_a=*/false, /*reuse_b=*/false);
  }

  // C/D layout: VGPR r -> M = r + 8*half, N = lane&15
#pragma unroll
  for (int r = 0; r < 8; ++r) {
    int row = m0 + r + half * 8;
    int col = n0 + l15;
    float v = c[r] + bias[col];
    if (RELU) v = fmaxf(v, 0.0f);
    out[(size_t)row * HID + col] = v;
  }
}

// ---- GCN propagate: out[row] += h[col] * dinv[row]*dinv[col] ---------------
// 16 threads per undirected edge; each handles a float4 feature chunk.
__global__ __launch_bounds__(256)
void propagate_kernel(const long long* __restrict__ e0,
                      const long long* __restrict__ e1,
                      const float* __restrict__ dinv,
                      const float* __restrict__ h,
                      float* __restrict__ out, long long E) {
  long long t = (long long)blockIdx.x * 256 + threadIdx.x;
  long long e = t >> 4;
  int f = (int)(t & 15) << 2;
  if (e >= 2 * E) return;
  bool fwd = e < E;
  long long idx = fwd ? e : e - E;
  long long r = fwd ? e0[idx] : e1[idx];
  long long c = fwd ? e1[idx] : e0[idx];
  float norm = dinv[r] * dinv[c];
  const float4 hv = *(const float4*)(h + c * HID + f);
  float* o = out + r * HID + f;
  unsafeAtomicAdd(o + 0, hv.x * norm);
  unsafeAtomicAdd(o + 1, hv.y * norm);
  unsafeAtomicAdd(o + 2, hv.z * norm);
  unsafeAtomicAdd(o + 3, hv.w * norm);
}

extern "C" void kernel_launch(void* const* d_in, const int* in_sizes, int n_in,
                              void* d_out, int out_size, void* d_ws, size_t ws_size,
                              hipStream_t stream) {
  const float*     x     = (const float*)d_in[0];
  const long long* ei    = (const long long*)d_in[1];
  const float*     W_in  = (const float*)d_in[2];
  const float*     b_in  = (const float*)d_in[3];
  const float*     W1    = (const float*)d_in[4];
  const float*     b1    = (const float*)d_in[5];
  const float*     W2    = (const float*)d_in[6];
  const float*     b2    = (const float*)d_in[7];
  const float*     W_out = (const float*)d_in[8];
  const float*     b_out = (const float*)d_in[9];
  float*           out   = (float*)d_out;

  const int       N  = in_sizes[0] / 128;     // 100000
  const long long E  = in_sizes[1] / 2;       // 1600000
  const long long* e0 = ei;
  const long long* e1 = ei + E;

  // workspace: hA [N*64] then deg/dinv [N]
  float* hA  = (float*)d_ws;
  float* deg = hA + (size_t)N * HID;

  // ---- degrees -> dinv ----
  hipMemsetAsync(deg, 0, (size_t)N * sizeof(float), stream);
  degree_kernel<<<(int)((E + 255) / 256), 256, 0, stream>>>(e0, e1, deg, E);
  dinv_kernel<<<(N + 255) / 256, 256, 0, stream>>>(deg, N);

  const int gblocks = N / 32;                 // 3125 (exact)
  const long long pthreads = 2 * E * 16;      // 51.2M (exact multiple of 256)
  const int pblocks = (int)((pthreads + 255) / 256);

  // h = x @ W_in + b_in
  gemm_bias_kernel<128, false><<<gblocks, 256, 0, stream>>>(x, W_in, b_in, hA);

  // layer 1: h = propagate(relu(h @ W1 + b1))
  gemm_bias_kernel<64, true><<<gblocks, 256, 0, stream>>>(hA, W1, b1, out);
  hipMemsetAsync(hA, 0, (size_t)N * HID * sizeof(float), stream);
  propagate_kernel<<<pblocks, 256, 0, stream>>>(e0, e1, deg, out, hA, E);

  // layer 2: h = propagate(relu(h @ W2 + b2))
  gemm_bias_kernel<64, true><<<gblocks, 256, 0, stream>>>(hA, W2, b2, out);
  hipMemsetAsync(hA, 0, (size_t)N * HID * sizeof(float), stream);
  propagate_kernel<<<pblocks, 256, 0, stream>>>(e0, e1, deg, out, hA, E);

  // out = h @ W_out + b_out
  gemm_bias_kernel<64, false><<<gblocks, 256, 0, stream>>>(hA, W_out, b_out, out);
}